// RBFPotentialPeriodic_77120432767507
// MI455X (gfx1250) — compile-verified
//
#include <hip/hip_runtime.h>

typedef __attribute__((ext_vector_type(2))) float v2f;
typedef __attribute__((ext_vector_type(8))) float v8f;

#define NN 8192
#define MM 1024
#define DD 32
#define MSPLIT 8                            // M-chunks per n-strip (occupancy)
#define TILES_PER_CHUNK (MM / 16 / MSPLIT)  // 8 tiles of 16 centers
#define PAD 68  // LDS row pitch in floats: 16B-aligned rows + conflict-free frags

// Kernel 1: CS[m][k] = (k<32 ? cos(c[m][k]) : sin(c[m][k-32])),
// inv_s2[m] = exp(-2*logsig[m]).
__global__ __launch_bounds__(256) void rbf_precompute(
    const float* __restrict__ c, const float* __restrict__ logsig,
    float* __restrict__ CS, float* __restrict__ inv_s2) {
  int t = blockIdx.x * blockDim.x + threadIdx.x;
  if (t < MM * 64) {
    int m = t >> 6;
    int k = t & 63;
    float v = c[m * DD + (k & (DD - 1))];
    CS[t] = (k < DD) ? __cosf(v) : __sinf(v);
  }
  if (t < MM) {
    inv_s2[t] = __expf(-2.0f * logsig[t]);
  }
}

// Kernel 2: one wave per (16-row n-strip, M-chunk). All 8 waves of a block
// share the same M-chunk, so the chunk's CS rows are staged once into LDS.
//   S = XS * CS^T via v_wmma_f32_16x16x4_f32 (K=64, two 8-deep acc chains),
//   Phi = exp((S - D) * inv_s2[m]) * w[m] accumulated in registers,
//   butterfly-reduced over tile columns, partial sum -> workspace.
__global__ __launch_bounds__(256) void rbf_wmma(
    const float* __restrict__ x, const float* __restrict__ CS,
    const float* __restrict__ inv_s2, const float* __restrict__ w,
    float* __restrict__ partial) {
  __shared__ float Bs[TILES_PER_CHUNK * 16 * PAD];  // 128 rows x 68 floats

  const int wave = threadIdx.x >> 5;
  const int lane = threadIdx.x & 31;
  const int nbase = (blockIdx.x * 8 + wave) * 16;
  const int chunk = blockIdx.y;
  const int col = lane & 15;  // A-matrix row for this lane; also D-tile column
  const int hi = lane >> 4;   // K-pair select: lanes 0-15 -> K{0,1}, 16-31 -> K{2,3}

  // Stage this chunk's 128 CS rows into LDS (pad 64 -> 68 floats per row).
  // 256 threads: 2 per row, each copies 32 floats as 8 x b128.
  {
    const int row = threadIdx.x >> 1;
    const int half = threadIdx.x & 1;
    const float* src =
        CS + (size_t)(chunk * TILES_PER_CHUNK * 16 + row) * 64 + half * 32;
    float* dst = &Bs[row * PAD + half * 32];
#pragma unroll
    for (int i = 0; i < 8; ++i)
      *(float4*)(dst + 4 * i) = *(const float4*)(src + 4 * i);
  }

  // A fragments: af[kk] = XS[nbase+col][4*kk + 2*hi + {0,1}] per the
  // 32-bit 16x4 A layout (VGPR0 = even K of pair, VGPR1 = odd K).
  v2f af[16];
  const float* xr = x + (size_t)(nbase + col) * DD;
#pragma unroll
  for (int kk = 0; kk < 16; ++kk) {
    int k0 = 4 * kk + 2 * hi;  // even; k0,k0+1 land in the same cos/sin half
    float x0 = xr[k0 & (DD - 1)];
    float x1 = xr[(k0 + 1) & (DD - 1)];
    v2f f;
    if (k0 < DD) {
      f.x = __cosf(x0);
      f.y = __cosf(x1);
    } else {
      f.x = __sinf(x0);
      f.y = __sinf(x1);
    }
    af[kk] = f;
  }

  __syncthreads();

  float acc[8];
#pragma unroll
  for (int j = 0; j < 8; ++j) acc[j] = 0.0f;

  for (int mtl = 0; mtl < TILES_PER_CHUNK; ++mtl) {
    const int m = (chunk * TILES_PER_CHUNK + mtl) * 16 + col;

    // Preload all 16 B fragments (ds_load_b64, conflict-free banks) so the
    // scheduler can clause them and wait once, then run two independent
    // 8-deep WMMA accumulation chains for ILP in the XDL pipe.
    const float* br = &Bs[(mtl * 16 + col) * PAD + 2 * hi];
    v2f bf[16];
#pragma unroll
    for (int kk = 0; kk < 16; ++kk) bf[kk] = *(const v2f*)(br + 4 * kk);

    v8f C0 = {}, C1 = {};
#pragma unroll
    for (int kk = 0; kk < 8; ++kk) {
      C0 = __builtin_amdgcn_wmma_f32_16x16x4_f32(false, af[kk], false, bf[kk],
                                                 (short)0, C0, false, false);
      C1 = __builtin_amdgcn_wmma_f32_16x16x4_f32(false, af[kk + 8], false,
                                                 bf[kk + 8], (short)0, C1,
                                                 false, false);
    }

    float wv = w[m];
    float is2 = inv_s2[m];
    // D-tile layout: VGPR j -> rows j / j+8 (lane halves), column = lane&15.
#pragma unroll
    for (int j = 0; j < 8; ++j)
      acc[j] += __expf((C0[j] + C1[j] - (float)DD) * is2) * wv;
  }

  // Reduce over the 16 tile columns (lanes within each 16-lane half).
#pragma unroll
  for (int j = 0; j < 8; ++j) {
#pragma unroll
    for (int off = 1; off < 16; off <<= 1)
      acc[j] += __shfl_xor(acc[j], off, 32);
  }

  // Lanes 0..7 own rows 0..7; lanes 16..23 own rows 8..15. One writer per
  // (n, chunk) -> deterministic.
  if (col < 8) {
    const int r = col + 8 * hi;
    float myacc = 0.0f;
#pragma unroll
    for (int j = 0; j < 8; ++j) myacc = (col == j) ? acc[j] : myacc;
    partial[(size_t)chunk * NN + nbase + r] = myacc;
  }
}

// Kernel 3: out[n] = sum_ch partial[ch][n] + x[n].a + b  (deterministic order)
__global__ __launch_bounds__(256) void rbf_finalize(
    const float* __restrict__ x, const float* __restrict__ a,
    const float* __restrict__ bsc, const float* __restrict__ partial,
    float* __restrict__ out) {
  int n = blockIdx.x * blockDim.x + threadIdx.x;
  if (n < NN) {
    float s = bsc[0];
#pragma unroll
    for (int ch = 0; ch < MSPLIT; ++ch) s += partial[(size_t)ch * NN + n];
    const float* xr = x + (size_t)n * DD;
    float xa = 0.0f;
#pragma unroll
    for (int d = 0; d < DD; ++d) xa += xr[d] * a[d];
    out[n] = s + xa;
  }
}

extern "C" void kernel_launch(void* const* d_in, const int* in_sizes, int n_in,
                              void* d_out, int out_size, void* d_ws,
                              size_t ws_size, hipStream_t stream) {
  const float* x = (const float*)d_in[0];
  const float* c = (const float*)d_in[1];
  const float* logsig = (const float*)d_in[2];
  const float* w = (const float*)d_in[3];
  const float* a = (const float*)d_in[4];
  const float* b = (const float*)d_in[5];
  float* out = (float*)d_out;

  float* CS = (float*)d_ws;      // MM*64 floats   = 256 KB
  float* inv_s2 = CS + MM * 64;  // MM floats      =   4 KB
  float* partial = inv_s2 + MM;  // MSPLIT*NN fl.  = 256 KB

  rbf_precompute<<<(MM * 64 + 255) / 256, 256, 0, stream>>>(c, logsig, CS,
                                                            inv_s2);
  rbf_wmma<<<dim3(NN / 16 / 8, MSPLIT), 256, 0, stream>>>(x, CS, inv_s2, w,
                                                          partial);
  rbf_finalize<<<(NN + 255) / 256, 256, 0, stream>>>(x, a, b, partial, out);
}